// SVD_AE_37701222924909
// MI455X (gfx1250) — compile-verified
//
#include <hip/hip_runtime.h>

// SVD-AE rating reconstruction for MI455X (gfx1250, wave32, WMMA).
// rating = (norm_adj @ item_sv) * (1/lambda) @ (user_sv^T @ adj_mat)
//
// U=10000, I=12000, RANK=64. All fp32; uses V_WMMA_F32_16X16X4_F32 to keep
// full fp32 precision while hitting the matrix pipes.

#define U_DIM 10000
#define I_DIM 12000
#define R_DIM 64

typedef __attribute__((ext_vector_type(2))) float v2f;
typedef __attribute__((ext_vector_type(8))) float v8f;

// D = A(16x4 f32) * B(4x16 f32) + C(16x16 f32)
static __device__ __forceinline__ v8f wmma4(v2f a, v2f b, v8f c) {
  return __builtin_amdgcn_wmma_f32_16x16x4_f32(
      /*neg_a=*/false, a, /*neg_b=*/false, b,
      /*c_mod=*/(short)0, c, /*reuse_a=*/false, /*reuse_b=*/false);
}

// ---------------------------------------------------------------------------
// Kernel 1: left[U,64] = norm_adj[U,I] @ item_sv[I,64]
// Grid: U/16 blocks x 128 threads (4 waves); wave w owns rank-column tile w.
// Streams norm_adj exactly once from HBM (A operand, b64/lane per k-step).
// ---------------------------------------------------------------------------
__global__ __launch_bounds__(128) void k_left(const float* __restrict__ norm_adj,
                                              const float* __restrict__ item_sv,
                                              float* __restrict__ left) {
  const int lane = threadIdx.x & 31;
  const int wave = threadIdx.x >> 5;      // 0..3 -> rank tile
  const int l16  = lane & 15;
  const int half = lane >> 4;             // 0/1
  const int m0   = blockIdx.x * 16;       // user rows
  const int n0   = wave * 16;             // rank cols

  v8f c = {};
  // A frag: lane holds norm_adj[m0+l16][k + 2*half + {0,1}]
  const float* aptr = norm_adj + (size_t)(m0 + l16) * I_DIM + 2 * half;
  // B frag: VGPR0 holds rows {k, k+2}, VGPR1 rows {k+1, k+3} at col n0+l16
  const float* bptr = item_sv + (size_t)(2 * half) * R_DIM + n0 + l16;

#pragma unroll 4
  for (int k = 0; k < I_DIM; k += 4) {
    v2f a; a.x = aptr[0]; a.y = aptr[1];
    v2f b; b.x = bptr[0]; b.y = bptr[R_DIM];
    c = wmma4(a, b, c);
    aptr += 4;
    bptr += 4 * R_DIM;
  }

#pragma unroll
  for (int r = 0; r < 8; ++r) {
    // C layout: VGPR r -> rows {r, r+8}, lanes give col n0+l16
    left[(size_t)(m0 + r + 8 * half) * R_DIM + n0 + l16] = c[r];
  }
}

// ---------------------------------------------------------------------------
// Kernel 2: right[64,I] = diag(1/lambda) * (user_sv^T[64,U] @ adj_mat[U,I])
// Grid: I/16 blocks x 128 threads (4 waves); wave w owns rank-row tile w.
// Streams adj_mat exactly once; 1/lambda folded into the store (scaling
// distributes over the contraction).
// ---------------------------------------------------------------------------
__global__ __launch_bounds__(128) void k_right(const float* __restrict__ user_sv,
                                               const float* __restrict__ adj,
                                               const float* __restrict__ lambda_mat,
                                               float* __restrict__ right) {
  const int lane = threadIdx.x & 31;
  const int wave = threadIdx.x >> 5;      // 0..3 -> rank-row tile
  const int l16  = lane & 15;
  const int half = lane >> 4;
  const int m0   = wave * 16;             // rank rows
  const int n0   = blockIdx.x * 16;       // item cols

  v8f c = {};
  // A[k][u] = user_sv[u][k]: lane holds user_sv[u + 2*half + {0,1}][m0+l16]
  const float* aptr = user_sv + (size_t)(2 * half) * R_DIM + m0 + l16;
  // B[u][i] = adj[u][i]
  const float* bptr = adj + (size_t)(2 * half) * I_DIM + n0 + l16;

#pragma unroll 4
  for (int u = 0; u < U_DIM; u += 4) {
    v2f a; a.x = aptr[0]; a.y = aptr[R_DIM];
    v2f b; b.x = bptr[0]; b.y = bptr[I_DIM];
    c = wmma4(a, b, c);
    aptr += 4 * R_DIM;
    bptr += 4 * (size_t)I_DIM;
  }

#pragma unroll
  for (int r = 0; r < 8; ++r) {
    const int kk = m0 + r + 8 * half;
    right[(size_t)kk * I_DIM + n0 + l16] = c[r] / lambda_mat[kk];
  }
}

// ---------------------------------------------------------------------------
// Kernel 3: rating[U,I] = left[U,64] @ right[64,I]
// Both factors are L2-resident (<6 MB); this kernel is HBM-store-bound.
// Grid: (I/16, ceil((U/16)/8)) x 256 threads; 8 waves/block share the same
// B tile (right columns) -> high L0/L2 hit rate on factor reads.
// ---------------------------------------------------------------------------
__global__ __launch_bounds__(256) void k_rating(const float* __restrict__ left,
                                                const float* __restrict__ right,
                                                float* __restrict__ out) {
  const int lane  = threadIdx.x & 31;
  const int wave  = threadIdx.x >> 5;     // 0..7
  const int l16   = lane & 15;
  const int half  = lane >> 4;
  const int mtile = blockIdx.y * 8 + wave;
  if (mtile >= U_DIM / 16) return;        // wave-uniform: EXEC stays all-1s
  const int m0 = mtile * 16;
  const int n0 = blockIdx.x * 16;

  v8f c = {};
  const float* aptr = left + (size_t)(m0 + l16) * R_DIM + 2 * half;
  const float* bptr = right + (size_t)(2 * half) * I_DIM + n0 + l16;

#pragma unroll
  for (int k = 0; k < R_DIM; k += 4) {
    v2f a; a.x = aptr[0]; a.y = aptr[1];
    v2f b; b.x = bptr[0]; b.y = bptr[I_DIM];
    c = wmma4(a, b, c);
    aptr += 4;
    bptr += 4 * (size_t)I_DIM;
  }

#pragma unroll
  for (int r = 0; r < 8; ++r)
    out[(size_t)(m0 + r + 8 * half) * I_DIM + n0 + l16] = c[r];
}

// ---------------------------------------------------------------------------
extern "C" void kernel_launch(void* const* d_in, const int* in_sizes, int n_in,
                              void* d_out, int out_size, void* d_ws, size_t ws_size,
                              hipStream_t stream) {
  const float* lambda_mat = (const float*)d_in[0];   // [64]
  const float* user_sv    = (const float*)d_in[1];   // [10000,64]
  const float* item_sv    = (const float*)d_in[2];   // [12000,64]
  const float* adj_mat    = (const float*)d_in[3];   // [10000,12000]
  const float* norm_adj   = (const float*)d_in[4];   // [10000,12000]
  float* out = (float*)d_out;                        // [10000,12000]

  float* left  = (float*)d_ws;                       // U*64 floats (2.56 MB)
  float* right = left + (size_t)U_DIM * R_DIM;       // 64*I floats (3.07 MB)

  // GEMM-1 and GEMM-2 are independent; GEMM-3 consumes both (stream order).
  k_left  <<<dim3(U_DIM / 16), dim3(128), 0, stream>>>(norm_adj, item_sv, left);
  k_right <<<dim3(I_DIM / 16), dim3(128), 0, stream>>>(user_sv, adj_mat, lambda_mat, right);
  k_rating<<<dim3(I_DIM / 16, (U_DIM / 16 + 7) / 8), dim3(256), 0, stream>>>(left, right, out);
}